// NeuralCDE_35184372089448
// MI455X (gfx1250) — compile-verified
//
#include <hip/hip_runtime.h>
#include <math.h>

// Neural CDE RK4 integrator for MI455X (gfx1250).
//
// Latency-bound sequential chain (2044 dependent 512x512x32 GEMMs -> ~35.6
// GFLOP total; HBM traffic ~67MB is microseconds at 23.3 TB/s). So: exact-fp32
// tensor ops (V_WMMA_F32_16X16X4_F32, 8 chained per K=32 GEMM), native
// v_tanh_f32, DPP row_xmask butterfly reduction (VALU, keeps the LDS pipe free
// for WMMA operand traffic), and 4 barriers/step.
// Each workgroup owns a 16-batch tile and integrates all 511 steps privately
// (batches independent -> no grid sync). 16 waves x 2 head-tiles cover N=512.

typedef __attribute__((ext_vector_type(2))) float v2f;
typedef __attribute__((ext_vector_type(8))) float v8f;

#define B_    512
#define L_    512
#define C_    16
#define H_    32
#define BT    16            // batch rows per workgroup (one WMMA M-tile)
#define NW    16            // waves per workgroup
#define TPB   (NW * 32)     // 512 threads
#define TPW   2             // head tiles per wave: 16 waves * 2 = 32 = H
#define PAD   34            // LDS row pad: 8B-aligned float2, no bank conflicts
#define NSTEP (L_ - 1)

__device__ __forceinline__ float fast_tanh(float x) {
#if __has_builtin(__builtin_amdgcn_tanhf)
    return __builtin_amdgcn_tanhf(x);          // native v_tanh_f32
#else
    float e = __builtin_amdgcn_exp2f(x * 2.88539008177793f);  // 2*log2(e)
    return 1.0f - 2.0f * __builtin_amdgcn_rcpf(e + 1.0f);
#endif
}

// lane-XOR add within 16-lane DPP rows: VALU (v_mov_b32_dpp row_xmask) instead
// of ds_bpermute -- keeps the reduction off the LDS pipe.
template <int MASK>
__device__ __forceinline__ float xor_add16(float v) {
    int p = __builtin_amdgcn_update_dpp(0, __float_as_int(v),
                                        0x160 + MASK /* ROW_XMASK */,
                                        0xf, 0xf, true);
    return v + __int_as_float(p);
}

__global__ __launch_bounds__(TPB)
void neural_cde_rk4(const float* __restrict__ X,    // (B, L, C)
                    const float* __restrict__ Wi,   // (H, C)
                    const float* __restrict__ bi,   // (H)
                    const float* __restrict__ Wv,   // (H*C, H)
                    const float* __restrict__ bv,   // (H*C)
                    float* __restrict__ out)        // (B, H)
{
    __shared__ __align__(16) float zsh[2][BT * PAD];   // z state, double-buffered
    __shared__ __align__(16) float ksh[2][BT * PAD];   // RK slope, double-buffered
    __shared__ __align__(16) float ksum[BT * PAD];     // k1 + 2k2 + 2k3
    __shared__ __align__(16) float dxs[2][3][BT * C_]; // dX/ds at s=0,.5,1 x 2 steps

    const int tid  = threadIdx.x;
    const int lane = tid & 31;
    const int wave = tid >> 5;
    const int half = lane >> 4;    // 0: lanes 0-15, 1: lanes 16-31
    const int lm   = lane & 15;
    const int b0   = blockIdx.x * BT;

    // ---- B-matrix fragments: Wv^T tiles, register-resident for the run ----
    // B layout (16x4 f32, mirrors ISA A-table): lane column n = lane&15,
    // .x/.y hold K rows {4kk+2*half, +1}.  B[k][n] = Wv[(h*16+n)*H + k].
    v2f   bfr[TPW][8];
    float bvv[TPW];
    #pragma unroll
    for (int t = 0; t < TPW; ++t) {
        const int h = wave * TPW + t;           // head index 0..31
        const int j = h * C_ + lm;              // Wv row
        bvv[t] = bv[j];
        #pragma unroll
        for (int kk = 0; kk < 8; ++kk) {
            const int kb = 4 * kk + 2 * half;
            bfr[t][kk].x = Wv[(size_t)j * H_ + kb + 0];
            bfr[t][kk].y = Wv[(size_t)j * H_ + kb + 1];
        }
    }

    // ---- z0 = X[:,0,:] @ Wi^T + bi ----
    {
        const int m = tid >> 5, h = tid & 31;
        float s = bi[h];
        #pragma unroll
        for (int c = 0; c < C_; ++c)
            s += X[(size_t)(b0 + m) * L_ * C_ + c] * Wi[h * C_ + c];
        zsh[0][m * PAD + h] = s;
    }
    // ---- prologue: stage dX/ds for step 0 (Hermite coeffs collapse to
    //      backward-diff combos: s=0 -> m0, s=.5 -> 1.25*d_i-0.25*d_{i-1},
    //      s=1 -> m1 = d_i; at i=0, m0 = m1 = d_0) ----
    if (tid < BT * C_) {
        const int m = tid >> 4, c = tid & 15;
        const size_t base = (size_t)(b0 + m) * L_ * C_ + c;
        const float d0 = X[base + C_] - X[base];
        dxs[0][0][tid] = d0;
        dxs[0][1][tid] = d0;
        dxs[0][2][tid] = d0;
    }
    __syncthreads();

    for (int i = 0; i < NSTEP; ++i) {
        const int cur = i & 1, nxt = cur ^ 1;

        // cache this step's z A-fragments in registers.
        // A layout (ISA 16x4 f32): lane row m = lane&15,
        // .x/.y hold K = 4kk + 2*half + {0,1}.
        v2f zfr[8];
        #pragma unroll
        for (int kk = 0; kk < 8; ++kk)
            zfr[kk] = *(const v2f*)&zsh[cur][lm * PAD + 4 * kk + 2 * half];

        #pragma unroll
        for (int st = 0; st < 4; ++st) {
            const float alpha = (st == 3) ? 1.0f : 0.5f;
            const int   sidx  = (st == 0) ? 0 : (st == 3 ? 2 : 1);

            // A fragments: z_eff = z + alpha * k_prev
            v2f afr[8];
            if (st == 0) {
                #pragma unroll
                for (int kk = 0; kk < 8; ++kk) afr[kk] = zfr[kk];
            } else {
                #pragma unroll
                for (int kk = 0; kk < 8; ++kk) {
                    const v2f kp = *(const v2f*)
                        &ksh[(st + 1) & 1][lm * PAD + 4 * kk + 2 * half];
                    afr[kk].x = zfr[kk].x + alpha * kp.x;
                    afr[kk].y = zfr[kk].y + alpha * kp.y;
                }
            }

            // overlap next step's dX/ds staging with stage-3 compute
            // (wave-uniform branch: waves 0-7 only; X rows are L2-resident)
            if (st == 3 && i + 1 < NSTEP && tid < BT * C_) {
                const int m = tid >> 4, c = tid & 15;
                const size_t base = (size_t)(b0 + m) * L_ * C_ + c
                                  + (size_t)(i + 1) * C_;
                const float xi    = X[base];
                const float dcur  = X[base + C_] - xi;
                const float dprev = xi - X[base - C_];
                dxs[nxt][0][tid] = dprev;
                dxs[nxt][1][tid] = 1.25f * dcur - 0.25f * dprev;
                dxs[nxt][2][tid] = dcur;
                if (m == 0 && i + 3 < L_)        // gfx1250 global_prefetch_b8
                    __builtin_prefetch(&X[base + 2 * C_], 0, 3);
            }

            // per-lane dX weights (C-layout: lane column c = lane&15,
            // VGPR r -> row m = r + 8*half)
            float dxr[8];
            #pragma unroll
            for (int r = 0; r < 8; ++r)
                dxr[r] = dxs[cur][sidx][(r + 8 * half) * C_ + lm];

            #pragma unroll
            for (int t = 0; t < TPW; ++t) {
                // K=32 GEMM tile via 8 chained exact-fp32 WMMAs
                v8f acc = {};
                #pragma unroll
                for (int kk = 0; kk < 8; ++kk)
                    acc = __builtin_amdgcn_wmma_f32_16x16x4_f32(
                              false, afr[kk], false, bfr[t][kk],
                              (short)0, acc, false, false);

                // vf = tanh(u + bv), weight by dX/ds
                float val[8];
                #pragma unroll
                for (int r = 0; r < 8; ++r)
                    val[r] = fast_tanh(acc[r] + bvv[t]) * dxr[r];

                // einsum over c: DPP row_xmask butterfly across the 16 lanes
                // of each half (pure VALU, LDS pipe stays free)
                #pragma unroll
                for (int r = 0; r < 8; ++r) {
                    val[r] = xor_add16<1>(val[r]);
                    val[r] = xor_add16<2>(val[r]);
                    val[r] = xor_add16<4>(val[r]);
                    val[r] = xor_add16<8>(val[r]);
                }

                // writer lane per half: k slope + RK4 accumulation; at the
                // final stage fold z_new = z + (k1+2k2+2k3+k4)/6 directly.
                if (lm == 0) {
                    const int h = wave * TPW + t;
                    #pragma unroll
                    for (int r = 0; r < 8; ++r) {
                        const int m = r + 8 * half;
                        const int o = m * PAD + h;
                        if (st == 0) {
                            ksh[0][o]  = val[r];
                            ksum[o]    = val[r];
                        } else if (st < 3) {
                            ksh[st & 1][o] = val[r];
                            ksum[o]       += 2.0f * val[r];
                        } else {
                            zsh[nxt][o] = zsh[cur][o]
                                        + (ksum[o] + val[r]) * (1.0f / 6.0f);
                        }
                    }
                }
            }
            __syncthreads();
        }
    }

    // ---- write zT (after NSTEP=511 steps, z lives in buffer NSTEP&1) ----
    {
        const int m = tid >> 5, h = tid & 31;
        out[(size_t)(b0 + m) * H_ + h] = zsh[NSTEP & 1][m * PAD + h];
    }
}

extern "C" void kernel_launch(void* const* d_in, const int* in_sizes, int n_in,
                              void* d_out, int out_size, void* d_ws, size_t ws_size,
                              hipStream_t stream) {
    (void)in_sizes; (void)n_in; (void)out_size; (void)d_ws; (void)ws_size;
    const float* X  = (const float*)d_in[0];   // (512, 512, 16)
    const float* Wi = (const float*)d_in[1];   // (32, 16)
    const float* bi = (const float*)d_in[2];   // (32)
    const float* Wv = (const float*)d_in[3];   // (512, 32)
    const float* bv = (const float*)d_in[4];   // (512)
    float* out = (float*)d_out;                // (512, 32)

    neural_cde_rk4<<<B_ / BT, TPB, 0, stream>>>(X, Wi, bi, Wv, bv, out);
}